// Model_90692529422873
// MI455X (gfx1250) — compile-verified
//
#include <hip/hip_runtime.h>
#include <math.h>

#define N 16384
#define D 64
#define BM 128          // rows of S per workgroup
#define BN 64           // cols of S per workgroup
#define LDA 68          // padded LDS row stride (floats): 16B-aligned rows, conflict-free b64 reads

__constant__ float kE2 = 7.38905609893065f;   // exp(1/T), T = 0.5

typedef __attribute__((ext_vector_type(2))) float v2f;
typedef __attribute__((ext_vector_type(8))) float v8f;

// Kernel 1: row-normalize X into workspace, zero colsum/num accumulators.
__global__ void __launch_bounds__(256)
normalize_rows(const float* __restrict__ x, float* __restrict__ xn,
               float* __restrict__ colsum, float* __restrict__ num) {
  int r = blockIdx.x * blockDim.x + threadIdx.x;
  if (r >= N) return;
  const float4* xr = (const float4*)(x + (size_t)r * D);
  float4 buf[D / 4];
  float s = 0.f;
#pragma unroll
  for (int i = 0; i < D / 4; ++i) {
    float4 v = xr[i];
    buf[i] = v;
    s += v.x * v.x + v.y * v.y + v.z * v.z + v.w * v.w;
  }
  // reference clamps |xi||xj| at 1e-8; per-row rsqrt clamp is equivalent for this data
  float inv = rsqrtf(fmaxf(s, 1e-24f));
  float4* out = (float4*)(xn + (size_t)r * D);
#pragma unroll
  for (int i = 0; i < D / 4; ++i) {
    float4 v = buf[i];
    out[i] = make_float4(v.x * inv, v.y * inv, v.z * inv, v.w * inv);
  }
  colsum[r] = 0.f;
  num[r] = 0.f;
}

// Kernel 2: fused S = Xn*Xn^T tile GEMM (WMMA f32 16x16x4), e = exp(2*S),
// column-sum reduction (LDS then one global atomic per column per block),
// and positive-pair extraction num[j] = e[j^1, j].
__global__ void __launch_bounds__(256)
sim_exp_kernel(const float* __restrict__ xn, float* __restrict__ colsum,
               float* __restrict__ num) {
  __shared__ float Xa[BM * LDA];
  __shared__ float Xb[BN * LDA];
  __shared__ float scol[BN];

  const int tid = threadIdx.x;
  const int m0 = blockIdx.y * BM;   // S row block
  const int n0 = blockIdx.x * BN;   // S col block

  // Cooperative stage: A-panel (BM x 64) and B-panel (BN x 64) into padded LDS.
  {
    const float4* srcA = (const float4*)(xn + (size_t)m0 * D);
#pragma unroll
    for (int t = 0; t < (BM * D / 4) / 256; ++t) {
      int idx = tid + 256 * t;
      int row = idx >> 4;                 // 16 float4 per 64-float row
      int c4 = idx & 15;
      *(float4*)(&Xa[row * LDA + c4 * 4]) = srcA[idx];
    }
    const float4* srcB = (const float4*)(xn + (size_t)n0 * D);
#pragma unroll
    for (int t = 0; t < (BN * D / 4) / 256; ++t) {
      int idx = tid + 256 * t;
      int row = idx >> 4;
      int c4 = idx & 15;
      *(float4*)(&Xb[row * LDA + c4 * 4]) = srcB[idx];
    }
  }
  if (tid < BN) scol[tid] = 0.f;
  __syncthreads();

  const int lane = tid & 31;
  const int w    = tid >> 5;     // 8 waves
  const int nt   = w & 3;        // wave's 16-col tile within BN
  const int mtB  = w >> 2;       // 0/1: interleaved M-tiles
  const int half = lane >> 4;    // lane 16..31 selects K+2 / M+8 halves
  const int l16  = lane & 15;

  // Hoist all 16 K-step B-fragments for this wave's column tile.
  // B (4x16 f32) layout: VGPRv, lane L: K = v + 2*(L>=16), Ncol = L%16  -> contiguous float2.
  v2f bfrag[16];
  const float* brow = &Xb[(nt * 16 + l16) * LDA + 2 * half];
#pragma unroll
  for (int s = 0; s < 16; ++s) bfrag[s] = *(const v2f*)(brow + 4 * s);

#pragma unroll
  for (int t = 0; t < 4; ++t) {
    const int mt = mtB + 2 * t;            // 0..7: 16-row tile within BM
    v8f acc = {0.f, 0.f, 0.f, 0.f, 0.f, 0.f, 0.f, 0.f};
    const float* arow = &Xa[(mt * 16 + l16) * LDA + 2 * half];
#pragma unroll
    for (int s = 0; s < 16; ++s) {
      // A (16x4 f32) layout: VGPRv, lane L: M = L%16, K = v + 2*(L>=16) -> contiguous float2.
      v2f afrag = *(const v2f*)(arow + 4 * s);
      acc = __builtin_amdgcn_wmma_f32_16x16x4_f32(
          false, afrag, false, bfrag[s], (short)0, acc, false, false);
    }
    // Epilogue: e = exp(2*sim); C/D layout: acc[v] -> row M = v + 8*half, col N = l16.
    const int gj   = n0 + nt * 16 + l16;   // global column
    const int prow = gj ^ 1;               // positive-pair partner row
    float sum8 = 0.f;
#pragma unroll
    for (int v = 0; v < 8; ++v) {
      const int gi = m0 + mt * 16 + v + 8 * half;
      float e = __expf(2.0f * acc[v]);     // sim / T, T = 0.5
      sum8 += e;
      if (gi == prow) num[gj] = e;         // computed exactly once globally
    }
    atomicAdd(&scol[nt * 16 + l16], sum8); // LDS ds_add_f32
  }
  __syncthreads();
  if (tid < BN) atomicAdd(&colsum[n0 + tid], scol[tid]);  // 1 global atomic / col / block
}

// Kernel 3: loss = -log(mean(num[j] / (colsum[j] - exp(2))))
__global__ void __launch_bounds__(256)
finalize_kernel(const float* __restrict__ colsum, const float* __restrict__ num,
                float* __restrict__ out) {
  __shared__ float red[256];
  float s = 0.f;
  for (int j = threadIdx.x; j < N; j += 256)
    s += num[j] / (colsum[j] - kE2);
  red[threadIdx.x] = s;
  __syncthreads();
  for (int off = 128; off > 0; off >>= 1) {
    if (threadIdx.x < off) red[threadIdx.x] += red[threadIdx.x + off];
    __syncthreads();
  }
  if (threadIdx.x == 0) out[0] = -logf(red[0] / (float)N);
}

extern "C" void kernel_launch(void* const* d_in, const int* in_sizes, int n_in,
                              void* d_out, int out_size, void* d_ws, size_t ws_size,
                              hipStream_t stream) {
  (void)in_sizes; (void)n_in; (void)out_size; (void)ws_size;
  const float* x = (const float*)d_in[0];
  float* xn     = (float*)d_ws;                 // N*D floats (4 MB)
  float* colsum = xn + (size_t)N * D;           // N floats
  float* num    = colsum + N;                   // N floats
  float* out    = (float*)d_out;

  normalize_rows<<<N / 256, 256, 0, stream>>>(x, xn, colsum, num);
  dim3 grid(N / BN, N / BM);                    // 256 x 128 workgroups
  sim_exp_kernel<<<grid, 256, 0, stream>>>(xn, colsum, num);
  finalize_kernel<<<1, 256, 0, stream>>>(colsum, num, out);
}